// MyModel_61933428412330
// MI455X (gfx1250) — compile-verified
//
#include <hip/hip_runtime.h>

// Problem constants (match reference)
#define B_        4
#define N_        2048
#define P_        ((long long)N_ * (N_ - 1) / 2)   // 2,096,128
#define BP_       ((long long)B_ * P_)             // 8,384,512
#define BANDS     (N_ / 16)                        // 128 bands of 16 rows
#define NBLOCKS   (B_ * BANDS)                     // 512
#define CUT2      25.0f
#define NTHREADS  256
#define NWAVES    (NTHREADS / 32)
#define WPR       (N_ / 16)                        // 128 u16 mask words per row
#define NWORDS    (16 * WPR)                       // 2048 words per band
#define WPT       (NWORDS / NTHREADS)              // 8 words per thread

typedef __attribute__((ext_vector_type(2))) float v2f;
typedef __attribute__((ext_vector_type(8))) float v8f;
typedef __attribute__((ext_vector_type(2))) long long v2ll;   // 128-bit store unit

// Compute the 16xN validity bitmask for rows [i0, i0+16) of batch b.
// Gram matrix via V_WMMA_F32_16X16X4_F32 (K=4: x,y,z,0), d2 = ni + nj - 2*dot.
// s_mask[r*WPR + t] bit k  <=>  pair (i0+r, 16*t+k) is a valid edge.
__device__ __forceinline__ void compute_band_mask(
    const float* __restrict__ x, int b, int i0,
    float* s_x, float* s_y, float* s_z, float* s_n,
    unsigned short* s_mask)
{
    const int tid = threadIdx.x;

    // Stage all N points of batch b (+ squared norms) into LDS.
    const float* xb = x + (size_t)b * N_ * 3;
    for (int idx = tid; idx < N_; idx += NTHREADS) {
        float px = xb[idx * 3 + 0];
        float py = xb[idx * 3 + 1];
        float pz = xb[idx * 3 + 2];
        s_x[idx] = px; s_y[idx] = py; s_z[idx] = pz;
        s_n[idx] = px * px + py * py + pz * pz;
    }
    for (int w = tid; w < NWORDS; w += NTHREADS) s_mask[w] = 0;
    __syncthreads();

    const int lane = tid & 31;
    const int wave = tid >> 5;
    const int half = lane >> 4;   // 0: K=0,1 (x,y)   1: K=2,3 (z,0)
    const int lr   = lane & 15;

    // A operand: 16x4 f32 tile of band rows (ISA layout: lanes 0-15 K0/K1, 16-31 K2/K3)
    v2f a;
    a.x = half ? s_z[i0 + lr] : s_x[i0 + lr];
    a.y = half ? 0.0f         : s_y[i0 + lr];
    const int rowBase = i0 + (half ? 8 : 0);  // C layout: VGPR v -> M = v (+8 for hi lanes)

    // Only column tiles at/after the diagonal tile can contain j > i.
    const int tStart = i0 >> 4;
    for (int t = tStart + wave; t < WPR; t += NWAVES) {
        const int j0 = t << 4;
        // B operand: 4x16 f32 tile of candidate columns (transposed-symmetric layout)
        v2f bb;
        bb.x = half ? s_z[j0 + lr] : s_x[j0 + lr];
        bb.y = half ? 0.0f         : s_y[j0 + lr];

        v8f c = {};
        c = __builtin_amdgcn_wmma_f32_16x16x4_f32(
                /*neg_a=*/false, a, /*neg_b=*/false, bb,
                /*c_mod=*/(short)0, c, /*reuse_a=*/false, /*reuse_b=*/false);

        const int   j  = j0 + lr;       // column for this lane (both halves)
        const float nj = s_n[j];
        #pragma unroll
        for (int v = 0; v < 8; ++v) {
            const int   i  = rowBase + v;
            const float d2 = s_n[i] + nj - 2.0f * c[v];
            const bool  ok = (d2 <= CUT2) && (j > i);
            unsigned mm = (unsigned)__ballot(ok);   // wave32: bit L = lane L
            if (lane == 0) {
                s_mask[v * WPR + t]       = (unsigned short)(mm & 0xFFFFu); // rows v
                s_mask[(v + 8) * WPR + t] = (unsigned short)(mm >> 16);     // rows v+8
            }
        }
    }
    __syncthreads();
}

__global__ void __launch_bounds__(NTHREADS)
edge_count_kernel(const float* __restrict__ x, unsigned* __restrict__ blk_cnt)
{
    __shared__ float s_x[N_], s_y[N_], s_z[N_], s_n[N_];
    __shared__ unsigned short s_mask[NWORDS];
    __shared__ unsigned s_cnt;

    const int bid  = blockIdx.x;
    const int b    = bid / BANDS;
    const int i0   = (bid % BANDS) * 16;
    if (threadIdx.x == 0) s_cnt = 0;

    compute_band_mask(x, b, i0, s_x, s_y, s_z, s_n, s_mask);

    unsigned local = 0;
    for (int w = threadIdx.x; w < NWORDS; w += NTHREADS)
        local += (unsigned)__popc((unsigned)s_mask[w]);
    atomicAdd(&s_cnt, local);
    __syncthreads();
    if (threadIdx.x == 0) blk_cnt[bid] = s_cnt;
}

__global__ void blk_scan_kernel(const unsigned* __restrict__ blk_cnt,
                                unsigned* __restrict__ blk_off)
{
    if (threadIdx.x == 0 && blockIdx.x == 0) {
        unsigned acc = 0;
        for (int k = 0; k < NBLOCKS; ++k) { blk_off[k] = acc; acc += blk_cnt[k]; }
    }
}

__global__ void __launch_bounds__(NTHREADS)
edge_scatter_kernel(const float* __restrict__ x,
                    const unsigned* __restrict__ blk_off,
                    long long* __restrict__ out)
{
    __shared__ float s_x[N_], s_y[N_], s_z[N_], s_n[N_];
    __shared__ unsigned short s_mask[NWORDS];
    __shared__ unsigned s_part[NTHREADS];

    const int bid = blockIdx.x;
    const int b   = bid / BANDS;
    const int i0  = (bid % BANDS) * 16;

    compute_band_mask(x, b, i0, s_x, s_y, s_z, s_n, s_mask);

    const int tid = threadIdx.x;
    const int fw0 = tid * WPT;           // consecutive words => flat (row-major) order
    unsigned local = 0;
    #pragma unroll
    for (int k = 0; k < WPT; ++k) local += (unsigned)__popc((unsigned)s_mask[fw0 + k]);
    s_part[tid] = local;
    __syncthreads();

    if (tid == 0) {                      // 256-entry serial exclusive scan (off critical path)
        unsigned acc = 0;
        for (int t = 0; t < NTHREADS; ++t) { unsigned v = s_part[t]; s_part[t] = acc; acc += v; }
    }
    __syncthreads();

    long long pos = (long long)blk_off[bid] + s_part[tid];
    const long long rowOfs = (long long)b * N_;
    #pragma unroll
    for (int k = 0; k < WPT; ++k) {
        const int fw    = fw0 + k;
        unsigned  w     = (unsigned)s_mask[fw];
        const int i     = i0 + (fw >> 7);        // fw / WPR
        const int jbase = (fw & (WPR - 1)) << 4; // (fw % WPR) * 16
        while (w) {
            const int bit = __builtin_ctz(w);
            w &= w - 1;
            out[pos]       = rowOfs + i;            // out[0, pos]
            out[BP_ + pos] = rowOfs + jbase + bit;  // out[1, pos]
            ++pos;
        }
    }
}

// Fill d_out with -1 using 128-bit stores (GLOBAL_STORE_B128); total is even so
// the vector loop covers everything, scalar tail kept for generality.
__global__ void fill_neg1_kernel(long long* __restrict__ out, long long total)
{
    const long long nvec   = total >> 1;                 // number of 16B chunks
    v2ll* __restrict__ o2  = (v2ll*)out;
    const v2ll pat         = { -1LL, -1LL };
    long long idx    = (long long)blockIdx.x * blockDim.x + threadIdx.x;
    long long stride = (long long)gridDim.x * blockDim.x;
    for (long long v = idx; v < nvec; v += stride) o2[v] = pat;
    // scalar tail (empty when total is even)
    for (long long s = (nvec << 1) + idx; s < total; s += stride) out[s] = -1LL;
}

extern "C" void kernel_launch(void* const* d_in, const int* in_sizes, int n_in,
                              void* d_out, int out_size, void* d_ws, size_t ws_size,
                              hipStream_t stream)
{
    const float* x   = (const float*)d_in[0];       // [B, N, 3] f32
    long long*   out = (long long*)d_out;           // [2, B*P] int64

    unsigned* blk_cnt = (unsigned*)d_ws;            // 512 u32
    unsigned* blk_off = blk_cnt + NBLOCKS;          // 512 u32

    const long long total = 2LL * BP_;

    // 1) Pad entire output with -1 (tail after compaction stays -1).
    fill_neg1_kernel<<<8192, 256, 0, stream>>>(out, total);

    // 2) Per-band valid-edge counts (WMMA Gram-based distance eval).
    edge_count_kernel<<<NBLOCKS, NTHREADS, 0, stream>>>(x, blk_cnt);

    // 3) Exclusive scan of block counts -> global output bases (ascending flat order).
    blk_scan_kernel<<<1, 32, 0, stream>>>(blk_cnt, blk_off);

    // 4) Recompute masks, rank in flat order, scatter edge indices.
    edge_scatter_kernel<<<NBLOCKS, NTHREADS, 0, stream>>>(x, blk_off, out);
}